// Noise_27539330302294
// MI455X (gfx1250) — compile-verified
//
#include <hip/hip_runtime.h>

// Noise / bilinear grid-sample with wrap addressing.
//   image : [16, 1024, 1024] f32   (64 MB -> L2-resident on MI455X, 192MB L2)
//   coords: [N, 16, 2]       f32   (streamed once, non-temporal)
//   out   : [N, 16]          f32   (streamed once, non-temporal)
//
// Memory-bound: ~256MB HBM traffic -> ~11us floor @ 23.3 TB/s. No matmul
// structure, so the optimization targets are NT cache hints (keep image in
// L2), wide coalesced VMEM, 32-bit saddr+voffset gather addressing, and
// high wave32 occupancy to hide L2 gather latency.

typedef __attribute__((ext_vector_type(4))) float v4f;

#define FEAT_MASK   15u     // FEATURES = 16
#define RES_MASK    1023    // RESOLUTION = 1024
#define ROW_SHIFT   10      // log2(1024)
#define PLANE_SHIFT 20      // log2(1024*1024)

__device__ __forceinline__ float bilerp_one(const float* __restrict__ image,
                                            unsigned f, float cx, float cy) {
    const float x   = cx - 0.5f;
    const float y   = cy - 0.5f;
    const float x0f = floorf(x);
    const float y0f = floorf(y);
    const float xw  = x - x0f;
    const float yw  = y - y0f;
    const unsigned x0 = ((unsigned)(int)x0f) & RES_MASK;   // -1 & 1023 == 1023 (jnp.mod)
    const unsigned x1 = (x0 + 1) & RES_MASK;
    const unsigned y0 = ((unsigned)(int)y0f) & RES_MASK;
    const unsigned y1 = (y0 + 1) & RES_MASK;

    // 32-bit element indices -> global_load_b32 vDst, vIdx, s[base] scale_offset
    const unsigned base = f << PLANE_SHIFT;
    const unsigned r0   = base | (y0 << ROW_SHIFT);
    const unsigned r1   = base | (y1 << ROW_SHIFT);

    const float i00 = image[r0 | x0];
    const float i01 = image[r0 | x1];
    const float i10 = image[r1 | x0];
    const float i11 = image[r1 | x1];

    const float i0 = fmaf(xw, i01 - i00, i00);   // lerp(a,b,w) = a + w*(b-a)
    const float i1 = fmaf(xw, i11 - i10, i10);
    return fmaf(yw, i1 - i0, i0);
}

__global__ __launch_bounds__(256)
void noise_bilerp_kernel(const float* __restrict__ image,
                         const v4f*  __restrict__ coords4,   // 2 samples per v4f
                         v4f*        __restrict__ out4,      // 4 results per store
                         int n_quads) {                      // groups of 4 samples
    const int tid = blockIdx.x * blockDim.x + threadIdx.x;
    if (tid >= n_quads) return;

    // Streaming operands: non-temporal so they never displace the image in L2.
    // Two global_load_b128 ... th:TH_LOAD_NT
    const v4f c01 = __builtin_nontemporal_load(&coords4[2 * tid]);
    const v4f c23 = __builtin_nontemporal_load(&coords4[2 * tid + 1]);

    const unsigned flat = (unsigned)tid << 2;   // flat sample index = n*16 + f
    v4f r;
    r.x = bilerp_one(image,  flat      & FEAT_MASK, c01.x, c01.y);
    r.y = bilerp_one(image, (flat + 1) & FEAT_MASK, c01.z, c01.w);
    r.z = bilerp_one(image, (flat + 2) & FEAT_MASK, c23.x, c23.y);
    r.w = bilerp_one(image, (flat + 3) & FEAT_MASK, c23.z, c23.w);

    // global_store_b128 ... th:TH_STORE_NT
    __builtin_nontemporal_store(r, &out4[tid]);
}

extern "C" void kernel_launch(void* const* d_in, const int* in_sizes, int n_in,
                              void* d_out, int out_size, void* d_ws, size_t ws_size,
                              hipStream_t stream) {
    const float* image  = (const float*)d_in[0];   // [16,1024,1024]
    const v4f*   coords = (const v4f*)d_in[1];     // [N,16,2] as packs of 2 samples
    v4f*         out    = (v4f*)d_out;             // [N,16]   as packs of 4 samples

    const int n_quads = out_size / 4;              // 16M samples -> 4M quads
    const int block = 256;                          // 8 wave32 waves per block
    const int grid  = (n_quads + block - 1) / block;

    noise_bilerp_kernel<<<grid, block, 0, stream>>>(image, coords, out, n_quads);
}